// LSTM_30142080483675
// MI455X (gfx1250) — compile-verified
//
#include <hip/hip_runtime.h>
#include <hip/hip_bf16.h>
#include <math.h>

#define VOCAB 50257
#define H     1024
#define C_DIM 1024
#define NB    4
#define BATCH 2048

typedef float v2f __attribute__((ext_vector_type(2)));
typedef float v8f __attribute__((ext_vector_type(8)));
typedef int   gi4 __attribute__((vector_size(16)));   // matches builtin param type

// ---------------------------------------------------------------------------
// gfx1250 async global->LDS copies (ASYNCcnt-tracked), with sync fallback.
// Signatures (from hipcc diagnostics): param0 = AS1 (global) typed pointer,
// param1 = AS3 (LDS) pointer, then imm offset, imm cpol.
// ---------------------------------------------------------------------------
#if defined(__has_builtin)
#  if __has_builtin(__builtin_amdgcn_global_load_async_to_lds_b128) && \
      __has_builtin(__builtin_amdgcn_global_load_async_to_lds_b32)
#    define USE_ASYNC_LDS 1
#  endif
#endif

__device__ __forceinline__ void lds_copy_b128(const float* g, float* l)
{
#ifdef USE_ASYNC_LDS
    __builtin_amdgcn_global_load_async_to_lds_b128(
        (__attribute__((address_space(1))) gi4*)(g),
        (__attribute__((address_space(3))) gi4*)(l), 0, 0);
#else
    *(float4*)l = *(const float4*)g;
#endif
}

__device__ __forceinline__ void lds_copy_b32(const float* g, float* l)
{
#ifdef USE_ASYNC_LDS
    __builtin_amdgcn_global_load_async_to_lds_b32(
        (__attribute__((address_space(1))) int*)(g),
        (__attribute__((address_space(3))) int*)(l), 0, 0);
#else
    *l = *g;
#endif
}

__device__ __forceinline__ void lds_copy_wait()
{
#ifdef USE_ASYNC_LDS
#  if __has_builtin(__builtin_amdgcn_s_wait_asynccnt)
    __builtin_amdgcn_s_wait_asynccnt(0);
#  else
    asm volatile("s_wait_asynccnt 0x0" ::: "memory");
#  endif
#endif
}

// ---------------------------------------------------------------------------
// fp32 WMMA GEMM: D[M,N] = concat_k(A1[:,0:K1], A2[:,0:K-K1]) @ W[K,N] + bias
// Tile 128x128, K-chunk 16, double-buffered async LDS staging.
// 256 threads = 8 waves; wave grid 4(m) x 2(n); each wave: 2x4 wmma tiles.
// LDS: A pitch 20 (conflict-free v2f A reads); B pair-interleaved
// [k/2][n][2] with pair pitch 288 (lane halves hit disjoint bank sets),
// so every B fragment is one aligned ds_load_b64.
// ---------------------------------------------------------------------------
#define APITCH 20
#define BPITCH 288   // floats per k-pair row (2*128 + 32)

__global__ __launch_bounds__(256)
void wmma_gemm_f32(const float* __restrict__ A1, const float* __restrict__ A2,
                   int K1, int K, int lda,
                   const float* __restrict__ W, const float* __restrict__ bias,
                   float* __restrict__ D, int N)
{
    __shared__ float As[2][128 * APITCH];  // 2 x 10.0 KB
    __shared__ float Bs[2][8 * BPITCH];    // 2 x  9.0 KB

    const int tid  = threadIdx.x;
    const int m0   = blockIdx.x * 128;
    const int n0   = blockIdx.y * 128;

    const int lane = tid & 31;
    const int w    = tid >> 5;
    const int wm   = (w >> 1) << 5;   // 0,32,64,96
    const int wn   = (w & 1) << 6;    // 0,64
    const int lh   = lane >> 4;       // lane half
    const int lm   = lane & 15;

    v8f acc[2][4];
    v8f zero = {0.f,0.f,0.f,0.f,0.f,0.f,0.f,0.f};
#pragma unroll
    for (int mt = 0; mt < 2; ++mt)
#pragma unroll
        for (int nt = 0; nt < 4; ++nt) acc[mt][nt] = zero;

    // staging decomposition (per thread):
    //   A: row = tid>>1 (0..127), 8 cols starting at (tid&1)*8 -> 2 x b128
    //   B: col = tid&127, k rows kr0..kr0+7 with kr0=(tid>>7)*8 -> 8 x b32
    const int ar  = tid >> 1;
    const int ac  = (tid & 1) * 8;
    const int bn  = tid & 127;
    const int bk0 = (tid >> 7) * 8;
    const int bcol = n0 + bn;
    const bool bok = (bcol < N);

    auto stage = [&](int buf, int kk) {
        const float* Ab; int kloc;
        if (kk < K1) { Ab = A1; kloc = kk; } else { Ab = A2; kloc = kk - K1; }
        const float* asrc = Ab + (size_t)(m0 + ar) * lda + kloc + ac;
        float* adst = &As[buf][ar * APITCH + ac];
#pragma unroll
        for (int j = 0; j < 2; ++j)
            lds_copy_b128(asrc + 4 * j, adst + 4 * j);
        if (bok) {
#pragma unroll
            for (int j = 0; j < 8; ++j) {
                const int kr = bk0 + j;
                lds_copy_b32(W + (size_t)(kk + kr) * N + bcol,
                             &Bs[buf][(kr >> 1) * BPITCH + bn * 2 + (kr & 1)]);
            }
        }
    };

    // prologue: stage chunk 0
    stage(0, 0);
    lds_copy_wait();
    __syncthreads();

    int buf = 0;
    for (int kk = 0; kk < K; kk += 16) {
        const int nxt = kk + 16;
        if (nxt < K) stage(buf ^ 1, nxt);   // async: streams while we compute

        // ---- compute chunk: 4 k-steps of V_WMMA_F32_16X16X4_F32
#pragma unroll
        for (int k0 = 0; k0 < 16; k0 += 4) {
            // A 16x4 layout: lane holds M=lm, K = k0 + 2*lh + {0,1}
            v2f a[2];
            a[0] = *(const v2f*)&As[buf][(wm + lm)      * APITCH + k0 + 2 * lh];
            a[1] = *(const v2f*)&As[buf][(wm + 16 + lm) * APITCH + k0 + 2 * lh];
            // B 4x16 layout: lane holds N=lm, same K pair -> one b64 each
            const int p = (k0 + 2 * lh) >> 1;
            v2f b[4];
#pragma unroll
            for (int nt = 0; nt < 4; ++nt)
                b[nt] = *(const v2f*)&Bs[buf][p * BPITCH + (wn + nt * 16 + lm) * 2];
#pragma unroll
            for (int mt = 0; mt < 2; ++mt)
#pragma unroll
                for (int nt = 0; nt < 4; ++nt)
                    acc[mt][nt] = __builtin_amdgcn_wmma_f32_16x16x4_f32(
                        false, a[mt], false, b[nt], (short)0, acc[mt][nt], false, false);
        }

        lds_copy_wait();   // my async loads into buf^1 have landed
        __syncthreads();   // everyone done reading buf / loading buf^1
        buf ^= 1;
    }

    // ---- epilogue: C/D layout: VGPR r -> M = r + 8*lh, lane lm -> N
#pragma unroll
    for (int nt = 0; nt < 4; ++nt) {
        const int n = n0 + wn + nt * 16 + lm;
        if (n < N) {
            const float bb = bias[n];
#pragma unroll
            for (int mt = 0; mt < 2; ++mt) {
#pragma unroll
                for (int r = 0; r < 8; ++r) {
                    const int m = m0 + wm + mt * 16 + 8 * lh + r;
                    D[(size_t)m * N + n] = acc[mt][nt][r] + bb;
                }
            }
        }
    }
}

// ---------------------------------------------------------------------------
// Embedding gather: xv[b,:] = emb[x[b],:]
// ---------------------------------------------------------------------------
__global__ __launch_bounds__(256)
void embed_kernel(const int* __restrict__ x, const float* __restrict__ emb,
                  float* __restrict__ xv)
{
    const int b = blockIdx.x;
    const int j = threadIdx.x * 4;             // H/4 = 256 float4 per row
    const int id = x[b];
    const float4 v = *(const float4*)(emb + (size_t)id * H + j);
    *(float4*)(xv + (size_t)b * H + j) = v;
}

// ---------------------------------------------------------------------------
// Cell update: c_new = sigmoid(gf)*c + sigmoid(gi)*tanh(gc); go <- sigmoid(go)
// ---------------------------------------------------------------------------
__global__ __launch_bounds__(256)
void cell_kernel(const float* __restrict__ gf, const float* __restrict__ gi,
                 const float* __restrict__ gc, float* __restrict__ go,
                 const float* __restrict__ c_in, float* __restrict__ c_out, int n)
{
    const int i = blockIdx.x * 256 + threadIdx.x;
    if (i < n) {
        const float f  = 1.f / (1.f + __expf(-gf[i]));
        const float ig = 1.f / (1.f + __expf(-gi[i]));
        const float ct = tanhf(gc[i]);
        const float o  = 1.f / (1.f + __expf(-go[i]));
        c_out[i] = f * c_in[i] + ig * ct;
        go[i] = o;
    }
}

// ---------------------------------------------------------------------------
// h_new = o * tanh(t), optional LayerNorm over the row (H=1024), 1 block/row.
// ---------------------------------------------------------------------------
__global__ __launch_bounds__(256)
void hnew_kernel(const float* __restrict__ t, const float* __restrict__ o,
                 float* __restrict__ h_out, int do_ln)
{
    __shared__ float rs[256];
    __shared__ float rs2[256];
    const int b   = blockIdx.x;
    const int tid = threadIdx.x;
    const float* tr  = t + (size_t)b * H;
    const float* orr = o + (size_t)b * H;

    float v[4];
    float s = 0.f, s2 = 0.f;
#pragma unroll
    for (int j = 0; j < 4; ++j) {
        const int col = tid + j * 256;
        const float hv = orr[col] * tanhf(tr[col]);
        v[j] = hv; s += hv; s2 += hv * hv;
    }

    if (do_ln) {
        rs[tid] = s; rs2[tid] = s2;
        __syncthreads();
        for (int off = 128; off > 0; off >>= 1) {
            if (tid < off) { rs[tid] += rs[tid + off]; rs2[tid] += rs2[tid + off]; }
            __syncthreads();
        }
        const float mu  = rs[0] * (1.f / H);
        const float var = rs2[0] * (1.f / H) - mu * mu;
        const float inv = rsqrtf(var + 1e-5f);
#pragma unroll
        for (int j = 0; j < 4; ++j) v[j] = (v[j] - mu) * inv;
    }

    float* hr = h_out + (size_t)b * H;
#pragma unroll
    for (int j = 0; j < 4; ++j) hr[tid + j * 256] = v[j];
}

// ---------------------------------------------------------------------------
extern "C" void kernel_launch(void* const* d_in, const int* in_sizes, int n_in,
                              void* d_out, int out_size, void* d_ws, size_t ws_size,
                              hipStream_t stream)
{
    const int*   x    = (const int*)  d_in[0];
    const float* hs   = (const float*)d_in[1];
    const float* cs   = (const float*)d_in[2];
    const float* emb  = (const float*)d_in[3];
    const float* Wf   = (const float*)d_in[4];
    const float* bf   = (const float*)d_in[5];
    const float* Wi   = (const float*)d_in[6];
    const float* bi   = (const float*)d_in[7];
    const float* Wc   = (const float*)d_in[8];
    const float* bc   = (const float*)d_in[9];
    const float* Wo   = (const float*)d_in[10];
    const float* bo   = (const float*)d_in[11];
    const float* Wch  = (const float*)d_in[12];
    const float* bch  = (const float*)d_in[13];
    const float* Wout = (const float*)d_in[14];
    const float* bout = (const float*)d_in[15];

    float* out        = (float*)d_out;
    float* out_logits = out;
    float* out_hs     = out + (size_t)BATCH * VOCAB;
    float* out_cs     = out_hs + (size_t)NB * BATCH * H;

    const size_t BH = (size_t)BATCH * H;
    float* ws = (float*)d_ws;
    float* xv = ws;            // [B,H]
    float* gf = xv + BH;
    float* gi = gf + BH;
    float* gc = gi + BH;
    float* go = gc + BH;
    float* tt = go + BH;       // [B,H]  (6*B*H floats = 48 MB total)

    // 1) embedding gather
    embed_kernel<<<BATCH, 256, 0, stream>>>(x, emb, xv);

    const dim3 gg(BATCH / 128, H / 128);   // 16 x 8
    for (int k = 0; k < NB; ++k) {
        const float* xin = (k == 0) ? xv : (out_hs + (size_t)(k - 1) * BH);
        const float* hk  = hs + (size_t)k * BH;
        const float* ck  = cs + (size_t)k * BH;
        const size_t wko = (size_t)k * 2 * H * H;

        // 2) four gate GEMMs: [xin ; h] @ W* + b*
        wmma_gemm_f32<<<gg, 256, 0, stream>>>(xin, hk, H, 2 * H, H, Wf + wko, bf + (size_t)k * H, gf, H);
        wmma_gemm_f32<<<gg, 256, 0, stream>>>(xin, hk, H, 2 * H, H, Wi + wko, bi + (size_t)k * H, gi, H);
        wmma_gemm_f32<<<gg, 256, 0, stream>>>(xin, hk, H, 2 * H, H, Wc + wko, bc + (size_t)k * H, gc, H);
        wmma_gemm_f32<<<gg, 256, 0, stream>>>(xin, hk, H, 2 * H, H, Wo + wko, bo + (size_t)k * H, go, H);

        // 3) cell state update (writes c_new directly into d_out's cs region)
        float* cnew = out_cs + (size_t)k * BH;
        cell_kernel<<<(BATCH * H) / 256, 256, 0, stream>>>(gf, gi, gc, go, ck, cnew, BATCH * H);

        // 4) t = c_new @ Wch + bch
        wmma_gemm_f32<<<gg, 256, 0, stream>>>(cnew, cnew, C_DIM, C_DIM, C_DIM,
                                              Wch + (size_t)k * C_DIM * H, bch + (size_t)k * H, tt, H);

        // 5) h_new = o * tanh(t) (+ LayerNorm except last block) -> d_out hs region
        hnew_kernel<<<BATCH, 256, 0, stream>>>(tt, go, out_hs + (size_t)k * BH, (k < NB - 1) ? 1 : 0);
    }

    // 6) logits = h_last @ Wout + bout   (N = 50257, tail-guarded)
    const float* hlast = out_hs + (size_t)(NB - 1) * BH;
    const dim3 gl(BATCH / 128, (VOCAB + 127) / 128);   // 16 x 393
    wmma_gemm_f32<<<gl, 256, 0, stream>>>(hlast, hlast, H, H, H, Wout, bout, out_logits, VOCAB);
}